// DA_conv_68118181314723
// MI455X (gfx1250) — compile-verified
//
#include <hip/hip_runtime.h>

typedef __attribute__((ext_vector_type(2))) float v2f;
typedef __attribute__((ext_vector_type(8))) float v8f;

#define C_ 64
#define H_ 128
#define W_ 128
#define KTAPS 9
#define LDS_STRIDE 132   // 128 + 4 pad to spread LDS banks

__device__ __forceinline__ float lrelu(float v) { return v >= 0.f ? v : 0.1f * v; }

// ---------------------------------------------------------------------------
// Kernel 1: kernel-generating MLP.  hid = lrelu(d @ Wk1^T); kern = hid @ Wk2^T
// One block per batch, 64 threads (thread t = hidden unit j, then channel c).
// Output: kernW[(b*64+c)*9 + tap]  (the per-(b,c) 3x3 taps), in d_ws.
// ---------------------------------------------------------------------------
__global__ void __launch_bounds__(64)
kgen_kernel(const float* __restrict__ d, const float* __restrict__ Wk1,
            const float* __restrict__ Wk2, float* __restrict__ kernW) {
  __shared__ float ds[64];
  __shared__ float hid[64];
  const int b = blockIdx.x;
  const int t = threadIdx.x;
  ds[t] = d[b * 64 + t];
  __syncthreads();
  float s = 0.f;
#pragma unroll
  for (int i = 0; i < 64; ++i) s += ds[i] * Wk1[t * 64 + i];
  hid[t] = lrelu(s);
  __syncthreads();
#pragma unroll
  for (int q = 0; q < KTAPS; ++q) {
    const int m = t * KTAPS + q;           // row of Wk2 (576 x 64)
    float s2 = 0.f;
    for (int j = 0; j < 64; ++j) s2 += hid[j] * Wk2[m * 64 + j];
    kernW[(b * 64 + t) * KTAPS + q] = s2;
  }
}

// ---------------------------------------------------------------------------
// Kernel 2: fused dynamic depthwise 3x3 conv + LeakyReLU (-> LDS) + 1x1 conv
// via V_WMMA_F32_16X16X4_F32.  One block = one (batch, row): 256 thr, 8 waves.
// ---------------------------------------------------------------------------
__global__ void __launch_bounds__(256)
fused_kernel(const float* __restrict__ x, const float* __restrict__ kernW,
             const float* __restrict__ Wc, const float* __restrict__ bc,
             float* __restrict__ out) {
  __shared__ float tmp[C_ * LDS_STRIDE];   // lrelu(dwconv) : [c][p]
  const int b = blockIdx.x >> 7;
  const int h = blockIdx.x & (H_ - 1);
  const int t = threadIdx.x;

  // -------- Phase A: depthwise 3x3 (SAME, zero pad) + lrelu into LDS -------
  {
    const int c   = t >> 2;              // 64 channels
    const int seg = (t & 3) * 32;        // 32-pixel segment
    float kv[KTAPS];
    const float* kp = kernW + (size_t)(b * 64 + c) * KTAPS;
#pragma unroll
    for (int j = 0; j < KTAPS; ++j) kv[j] = kp[j];

    const float* plane = x + ((size_t)(b * 64 + c) << 14);   // 128*128 plane
    const bool rv0 = (h > 0), rv2 = (h < H_ - 1);
    const float* r0 = plane + (size_t)(h - 1) * W_;
    const float* r1 = plane + (size_t)h * W_;
    const float* r2 = plane + (size_t)(h + 1) * W_;

    // 3x3 sliding window over columns seg-1 .. seg+32
    const bool cl = (seg > 0);           // left column in range?
    float w00 = (rv0 && cl) ? r0[seg - 1] : 0.f;
    float w01 = rv0 ? r0[seg]     : 0.f;
    float w02 = rv0 ? r0[seg + 1] : 0.f;
    float w10 = cl  ? r1[seg - 1] : 0.f;
    float w11 =       r1[seg];
    float w12 =       r1[seg + 1];
    float w20 = (rv2 && cl) ? r2[seg - 1] : 0.f;
    float w21 = rv2 ? r2[seg]     : 0.f;
    float w22 = rv2 ? r2[seg + 1] : 0.f;

    float* trow = tmp + c * LDS_STRIDE;
#pragma unroll 4
    for (int i = 0; i < 32; ++i) {
      float s = w00 * kv[0] + w01 * kv[1] + w02 * kv[2]
              + w10 * kv[3] + w11 * kv[4] + w12 * kv[5]
              + w20 * kv[6] + w21 * kv[7] + w22 * kv[8];
      trow[seg + i] = lrelu(s);
      const int nc = seg + i + 2;
      const bool cv = (nc < W_);
      w00 = w01; w01 = w02; w02 = (rv0 && cv) ? r0[nc] : 0.f;
      w10 = w11; w11 = w12; w12 = cv          ? r1[nc] : 0.f;
      w20 = w21; w21 = w22; w22 = (rv2 && cv) ? r2[nc] : 0.f;
    }
  }
  __syncthreads();

  // -------- Phase B: 1x1 conv GEMM  out[o,p] = Wc[o,:] . tmp[:,p] + bc -----
  // Wave w owns pixel tile p0 = 16*w; loops 4 o-tiles x 16 k-steps (K=4).
  const int wave  = t >> 5;
  const int lane  = t & 31;
  const int p0    = wave * 16;
  const int lh    = lane & 15;
  const int khi   = (lane >> 4) << 1;   // A/B K-offset: 0 (lo half) / 2 (hi)
  const int ohalf = (lane >> 4) << 3;   // C/D M-offset: 0 / 8

#pragma unroll
  for (int ot = 0; ot < 4; ++ot) {
    const int o0 = ot * 16;
    v8f acc;
#pragma unroll
    for (int v = 0; v < 8; ++v) acc[v] = bc[o0 + ohalf + v];  // fold bias in

    const float* arow = Wc + (size_t)(o0 + lh) * 64;          // A row = Wc[o]
#pragma unroll
    for (int kk = 0; kk < 16; ++kk) {
      const int c0 = kk * 4;
      v2f a, bf;
      // A 16x4: lane<16 -> K=0,1 ; lane>=16 -> K=2,3 (M = lane%16)
      a.x = arow[c0 + khi];
      a.y = arow[c0 + khi + 1];
      // B 4x16: VGPR0 row K=khi, VGPR1 row K=khi+1 (N = lane%16)
      bf.x = tmp[(c0 + khi)     * LDS_STRIDE + p0 + lh];
      bf.y = tmp[(c0 + khi + 1) * LDS_STRIDE + p0 + lh];
      acc = __builtin_amdgcn_wmma_f32_16x16x4_f32(
          /*neg_a=*/false, a, /*neg_b=*/false, bf,
          /*c_mod=*/(short)0, acc, /*reuse_a=*/false, /*reuse_b=*/false);
    }

    // D 16x16: VGPR v -> row o0+ohalf+v, col p0 + lane%16
    float* op = out + ((((size_t)b * C_ + o0 + ohalf) * H_ + h) * W_) + p0 + lh;
#pragma unroll
    for (int v = 0; v < 8; ++v) op[(size_t)v * H_ * W_] = acc[v];
  }
}

// ---------------------------------------------------------------------------
extern "C" void kernel_launch(void* const* d_in, const int* in_sizes, int n_in,
                              void* d_out, int out_size, void* d_ws, size_t ws_size,
                              hipStream_t stream) {
  const float* x   = (const float*)d_in[0];   // (16,64,128,128)
  const float* d   = (const float*)d_in[1];   // (16,64)
  const float* Wk1 = (const float*)d_in[2];   // (64,64)
  const float* Wk2 = (const float*)d_in[3];   // (576,64)
  const float* Wc  = (const float*)d_in[4];   // (64,64)
  const float* bc  = (const float*)d_in[5];   // (64,)
  float* out   = (float*)d_out;               // (16,64,128,128)
  float* kernW = (float*)d_ws;                // 16*64*9 floats = 36 KB scratch

  kgen_kernel<<<16, 64, 0, stream>>>(d, Wk1, Wk2, kernW);
  fused_kernel<<<16 * H_, 256, 0, stream>>>(x, kernW, Wc, bc, out);
}